// DGCN_12206297055753
// MI455X (gfx1250) — compile-verified
//
#include <hip/hip_runtime.h>

// Dims from reference: C=128 channels, N=1024 nodes, T=256 time, D=2 steps
#define C_   128
#define N_   1024
#define T_   256
#define NT_  (N_ * T_)        // 262144
#define TOPK_ 819             // int(1024 * 0.8)

typedef __attribute__((ext_vector_type(2))) float v2f;
typedef __attribute__((ext_vector_type(8))) float v8f;
typedef __attribute__((ext_vector_type(4))) unsigned int u32x4;
typedef __attribute__((ext_vector_type(4))) int i32x4;
typedef __attribute__((ext_vector_type(8))) int i32x8;

// F32 WMMA 16x16x4: A = 16x4 (v2f/lane), B = 4x16 (v2f/lane), C/D = 16x16 f32 (v8f/lane).
__device__ __forceinline__ v8f wmma4(v2f a, v2f b, v8f c) {
  return __builtin_amdgcn_wmma_f32_16x16x4_f32(false, a, false, b, (short)0, c,
                                               false, false);
}

// ---------------------------------------------------------------------------
// TDM: DMA a 2D f32 tile (tile_w contiguous elems per row, tile_h rows,
// row stride in elements) from global into LDS. Optional LDS padding of
// 8 DWORDs every 128 DWORDs (pad_interval=6 -> 128 dw, pad_amount=7 -> 8 dw)
// so a 128-wide slab gets LDS row stride 136 (bank-conflict-free half-waves).
// D# packing per CDNA5 ISA 8.3/8.4: g0 = {count|flags, lds_addr, gaddr_lo,
// gaddr_hi|type=2}; g1 = {mask/dsize/pad, dims, tile dims, stride0}.
// ---------------------------------------------------------------------------
__device__ __forceinline__ void tdm_load_2d_f32(void* lds_dst,
                                                const float* gsrc,
                                                unsigned tile_w, unsigned tile_h,
                                                unsigned row_stride_elems,
                                                bool pad128) {
  unsigned long long ga = (unsigned long long)(uintptr_t)gsrc;
  u32x4 g0;
  g0.x = 1u;                                    // count=1 (valid), no gather
  g0.y = (unsigned)(uintptr_t)lds_dst;          // LDS byte offset (addr[31:0])
  g0.z = (unsigned)(ga & 0xffffffffull);        // global_addr[31:0]
  g0.w = (unsigned)((ga >> 32) & 0x01ffffffu) | 0x80000000u;  // [56:32] | type=2
  i32x8 g1;
  unsigned d0 = (2u << 16);                     // data_size = 4 bytes
  if (pad128) d0 |= (1u << 20) | (6u << 22) | (7u << 25);  // pad 8dw / 128dw
  g1[0] = (int)d0;
  g1[1] = (int)((tile_w & 0xffffu) << 16);                     // tensor_dim0 lo
  g1[2] = (int)((tile_w >> 16) | ((tile_h & 0xffffu) << 16));  // d0 hi | d1 lo
  g1[3] = (int)((tile_h >> 16) | ((tile_w & 0xffffu) << 16));  // d1 hi | tile_dim0
  g1[4] = (int)(tile_h & 0xffffu);                             // tile_dim1 (dim2=0)
  g1[5] = (int)row_stride_elems;                               // dim0_stride lo32
  g1[6] = 0;
  g1[7] = 0;
  i32x4 z4 = {0, 0, 0, 0};
#if __has_include(<hip/amd_detail/amd_gfx1250_TDM.h>)
  i32x8 z8 = {0, 0, 0, 0, 0, 0, 0, 0};
  __builtin_amdgcn_tensor_load_to_lds(g0, g1, z4, z4, z8, 0);
#else
  __builtin_amdgcn_tensor_load_to_lds(g0, g1, z4, z4, 0);
#endif
}

#define SLAB_STRIDE 136  // 128 + 8 pad DWORDs per row

// ---------------------------------------------------------------------------
// K1: xc[o,j] = sum_c conv_w[o,c] * x[c,j] + conv_b[o]   (j = n*T + t)
// Block: 128-column j-slab. TDM stages x[0..127][j-slab] into padded LDS
// while all threads transpose W into LDS; inner loop is pure LDS + WMMA.
// Dynamic LDS: Wt 16384 f + Xs 128*136 f = 132 KB.
// ---------------------------------------------------------------------------
__global__ __launch_bounds__(256) void k_conv(const float* __restrict__ x,
                                              const float* __restrict__ W,
                                              const float* __restrict__ bias,
                                              float* __restrict__ xc) {
  extern __shared__ float smem[];
  float* Wt = smem;              // [64][256] paired-transposed W
  float* Xs = smem + 128 * 128;  // [128][SLAB_STRIDE] x slab
  const int tid = threadIdx.x;
  if (tid < 32)
    tdm_load_2d_f32(Xs, x + (size_t)blockIdx.x * 128, 128, 128, NT_, true);
  for (int i = tid; i < 128 * 128; i += 256) {
    int o = i >> 7, c = i & 127;
    Wt[(c >> 1) * 256 + o * 2 + (c & 1)] = W[i];
  }
  if (tid < 32) __builtin_amdgcn_s_wait_tensorcnt(0);
  __syncthreads();
  const int wave = tid >> 5, lane = tid & 31;
  const int l = lane & 15;
  const int koff = (lane >> 4) << 1;
  const int wl = (wave << 4) + l;   // column within slab
  const int j0 = blockIdx.x * 128 + (wave << 4);
  v8f acc[8] = {};
  for (int kb = 0; kb < 128; kb += 4) {
    const int kk = kb + koff;
    v2f b;
    b.x = Xs[kk * SLAB_STRIDE + wl];
    b.y = Xs[(kk + 1) * SLAB_STRIDE + wl];
    const float* wrow = &Wt[(kk >> 1) * 256];
#pragma unroll
    for (int ot = 0; ot < 8; ++ot) {
      v2f a = *(const v2f*)&wrow[(ot * 16 + l) * 2];
      acc[ot] = wmma4(a, b, acc[ot]);
    }
  }
  const int rbase = (lane >> 4) << 3;
#pragma unroll
  for (int ot = 0; ot < 8; ++ot) {
#pragma unroll
    for (int r = 0; r < 8; ++r) {
      int o = ot * 16 + rbase + r;
      xc[(size_t)o * NT_ + j0 + l] = acc[ot][r] + bias[o];
    }
  }
}

// ---------------------------------------------------------------------------
// K2: xcs[c,n] = sum_t xc[c,n,t]. One wave per row (row = c*N + n).
// ---------------------------------------------------------------------------
__global__ __launch_bounds__(256) void k_sum_t(const float* __restrict__ xc,
                                               float* __restrict__ xcs) {
  const int wave = threadIdx.x >> 5, lane = threadIdx.x & 31;
  const int row = blockIdx.x * 8 + wave;
  float s = 0.f;
#pragma unroll
  for (int j = 0; j < T_ / 32; ++j) s += xc[(size_t)row * T_ + lane + 32 * j];
#pragma unroll
  for (int off = 16; off > 0; off >>= 1) s += __shfl_xor(s, off, 32);
  if (lane == 0) xcs[row] = s;
}

// ---------------------------------------------------------------------------
// K3: ap1[n,m] = relu(scale * sum_c xcs[c,n]*memory[c,m])
//     ap2[n,m] = relu(scale * sum_c xcs[c,n]*xcs[c,m])
// One wave per 16x16 tile, WMMA with K=128.
// ---------------------------------------------------------------------------
__global__ __launch_bounds__(128) void k_adj_pre(const float* __restrict__ xcs,
                                                 const float* __restrict__ mem,
                                                 float* __restrict__ ap1,
                                                 float* __restrict__ ap2) {
  const int wave = threadIdx.x >> 5, lane = threadIdx.x & 31;
  const int tile = blockIdx.x * 4 + wave;
  const int n0 = (tile >> 6) * 16;
  const int m0 = (tile & 63) * 16;
  const int l = lane & 15, koff = (lane >> 4) << 1;
  v8f acc1 = {}, acc2 = {};
  for (int kb = 0; kb < 128; kb += 4) {
    const int kk = kb + koff;
    v2f a, b1, b2;
    a.x = xcs[kk * N_ + n0 + l];
    a.y = xcs[(kk + 1) * N_ + n0 + l];
    b1.x = mem[kk * N_ + m0 + l];
    b1.y = mem[(kk + 1) * N_ + m0 + l];
    b2.x = xcs[kk * N_ + m0 + l];
    b2.y = xcs[(kk + 1) * N_ + m0 + l];
    acc1 = wmma4(a, b1, acc1);
    acc2 = wmma4(a, b2, acc2);
  }
  const float scale = 0.03125f;  // 1/sqrt(1024)
  const int rbase = (lane >> 4) << 3;
#pragma unroll
  for (int r = 0; r < 8; ++r) {
    int n = n0 + rbase + r;
    float v1 = acc1[r] * scale;
    float v2 = acc2[r] * scale;
    ap1[(size_t)n * N_ + m0 + l] = v1 > 0.f ? v1 : 0.f;
    ap2[(size_t)n * N_ + m0 + l] = v2 > 0.f ? v2 : 0.f;
  }
}

// ---------------------------------------------------------------------------
// K4: per-row: af = softmax(w0*softmax(ap1) + w1*softmax(ap2) + b),
//     top-k(819) mask, adj = af * mask. One wave32 per row, all in registers;
//     exact k-th largest via 31-step bisection on positive-float bit order.
// ---------------------------------------------------------------------------
__global__ __launch_bounds__(32) void k_adj_row(const float* __restrict__ ap1,
                                                const float* __restrict__ ap2,
                                                const float* __restrict__ fc_w,
                                                const float* __restrict__ fc_b,
                                                float* __restrict__ adj) {
  const int row = blockIdx.x;
  const int lane = threadIdx.x;
  float v[32], z[32];

  {  // z = fc_w[0] * softmax(ap1_row)
#pragma unroll
    for (int i = 0; i < 32; ++i) v[i] = ap1[(size_t)row * N_ + lane + 32 * i];
    float mx = v[0];
#pragma unroll
    for (int i = 1; i < 32; ++i) mx = fmaxf(mx, v[i]);
#pragma unroll
    for (int off = 16; off > 0; off >>= 1) mx = fmaxf(mx, __shfl_xor(mx, off, 32));
    float s = 0.f;
#pragma unroll
    for (int i = 0; i < 32; ++i) { v[i] = __expf(v[i] - mx); s += v[i]; }
#pragma unroll
    for (int off = 16; off > 0; off >>= 1) s += __shfl_xor(s, off, 32);
    const float w0 = fc_w[0] / s;
#pragma unroll
    for (int i = 0; i < 32; ++i) z[i] = w0 * v[i];
  }
  {  // z += fc_w[1] * softmax(ap2_row) + fc_b
#pragma unroll
    for (int i = 0; i < 32; ++i) v[i] = ap2[(size_t)row * N_ + lane + 32 * i];
    float mx = v[0];
#pragma unroll
    for (int i = 1; i < 32; ++i) mx = fmaxf(mx, v[i]);
#pragma unroll
    for (int off = 16; off > 0; off >>= 1) mx = fmaxf(mx, __shfl_xor(mx, off, 32));
    float s = 0.f;
#pragma unroll
    for (int i = 0; i < 32; ++i) { v[i] = __expf(v[i] - mx); s += v[i]; }
#pragma unroll
    for (int off = 16; off > 0; off >>= 1) s += __shfl_xor(s, off, 32);
    const float w1 = fc_w[1] / s;
    const float bb = fc_b[0];
#pragma unroll
    for (int i = 0; i < 32; ++i) z[i] = z[i] + w1 * v[i] + bb;
  }
  {  // v = softmax(z)
    float mx = z[0];
#pragma unroll
    for (int i = 1; i < 32; ++i) mx = fmaxf(mx, z[i]);
#pragma unroll
    for (int off = 16; off > 0; off >>= 1) mx = fmaxf(mx, __shfl_xor(mx, off, 32));
    float s = 0.f;
#pragma unroll
    for (int i = 0; i < 32; ++i) { v[i] = __expf(z[i] - mx); s += v[i]; }
#pragma unroll
    for (int off = 16; off > 0; off >>= 1) s += __shfl_xor(s, off, 32);
    const float inv = 1.f / s;
#pragma unroll
    for (int i = 0; i < 32; ++i) v[i] *= inv;
  }
  // k-th largest threshold: bisection on bit pattern (values > 0)
  unsigned u = 0u;
  for (int bit = 30; bit >= 0; --bit) {
    const float tau = __uint_as_float(u | (1u << bit));
    int cnt = 0;
#pragma unroll
    for (int i = 0; i < 32; ++i) cnt += (v[i] >= tau) ? 1 : 0;
#pragma unroll
    for (int off = 16; off > 0; off >>= 1) cnt += __shfl_xor(cnt, off, 32);
    if (cnt >= TOPK_) u |= (1u << bit);
  }
  const float tau = __uint_as_float(u);
#pragma unroll
  for (int i = 0; i < 32; ++i)
    adj[(size_t)row * N_ + lane + 32 * i] = (v[i] >= tau) ? v[i] : 0.f;
}

// ---------------------------------------------------------------------------
// K5: out[c,m,t] = sum_n in[c,n,t] * adj[n,m]
// Block = (c, 16-wide t-tile). TDM stages B slab in[c][0..1023][t0..t0+15]
// (64 KB, 1024 rows x 16 f32) into LDS; 8 waves x 8 m-tiles cover all 1024 m.
// adj (4 MB) stays resident in the 192 MB L2.
// ---------------------------------------------------------------------------
__global__ __launch_bounds__(256) void k_diffuse(const float* __restrict__ in,
                                                 const float* __restrict__ adj,
                                                 float* __restrict__ out) {
  __shared__ __align__(16) float Bl[N_ * 16];
  const int c = blockIdx.x >> 4;
  const int t0 = (blockIdx.x & 15) << 4;
  const int tid = threadIdx.x;
  if (tid < 32) {
    tdm_load_2d_f32(Bl, in + (size_t)c * NT_ + t0, 16, N_, T_, false);
    __builtin_amdgcn_s_wait_tensorcnt(0);
  }
  __syncthreads();
  const int wave = tid >> 5, lane = tid & 31;
  const int l = lane & 15, koff = (lane >> 4) << 1;
  // warm L2/L0 for this wave's adj columns
  __builtin_prefetch(&adj[(size_t)((wave * 8) << 4) + l], 0, 3);
#pragma unroll 1
  for (int mt = 0; mt < 8; ++mt) {
    const int m0 = (wave * 8 + mt) << 4;
    v8f acc = {};
    for (int kb = 0; kb < N_; kb += 4) {
      const int kk = kb + koff;
      v2f a, b;
      a.x = adj[(size_t)kk * N_ + m0 + l];
      a.y = adj[(size_t)(kk + 1) * N_ + m0 + l];
      b.x = Bl[kk * 16 + l];
      b.y = Bl[(kk + 1) * 16 + l];
      acc = wmma4(a, b, acc);
    }
    const int rbase = (lane >> 4) << 3;
#pragma unroll
    for (int r = 0; r < 8; ++r) {
      int m = m0 + rbase + r;
      out[(size_t)c * NT_ + (size_t)m * T_ + t0 + l] = acc[r];
    }
  }
}

// ---------------------------------------------------------------------------
// K6: out[o,j] = (sum_c gw[o,c]*cur1[c,j] + sum_c gw[o,128+c]*cur2[c,j]
//                 + gb[o]) * emb[o,j] + skip[o,j]
// Two phases; each phase TDM-stages the 128x128 activation slab (padded) and
// re-stages the corresponding half of gw (transposed) into LDS.
// ---------------------------------------------------------------------------
__global__ __launch_bounds__(256) void k_gcn_out(const float* __restrict__ cur1,
                                                 const float* __restrict__ cur2,
                                                 const float* __restrict__ gw,
                                                 const float* __restrict__ gb,
                                                 const float* __restrict__ emb,
                                                 const float* __restrict__ skip,
                                                 float* __restrict__ out) {
  extern __shared__ float smem[];
  float* Wt = smem;              // [64][256] paired-transposed gw half
  float* Cs = smem + 128 * 128;  // [128][SLAB_STRIDE] activation slab
  const int tid = threadIdx.x;
  const int wave = tid >> 5, lane = tid & 31;
  const int l = lane & 15, koff = (lane >> 4) << 1;
  const int wl = (wave << 4) + l;
  const int j0 = blockIdx.x * 128 + (wave << 4);
  v8f acc[8] = {};
  const float* srcs[2] = {cur1, cur2};
#pragma unroll 1
  for (int ph = 0; ph < 2; ++ph) {
    __syncthreads();
    if (tid < 32)
      tdm_load_2d_f32(Cs, srcs[ph] + (size_t)blockIdx.x * 128, 128, 128, NT_,
                      true);
    for (int i = tid; i < 128 * 128; i += 256) {
      int o = i >> 7, cc = i & 127;
      Wt[(cc >> 1) * 256 + o * 2 + (cc & 1)] = gw[o * 256 + ph * 128 + cc];
    }
    if (tid < 32) __builtin_amdgcn_s_wait_tensorcnt(0);
    __syncthreads();
    for (int kb = 0; kb < 128; kb += 4) {
      const int kk = kb + koff;
      v2f b;
      b.x = Cs[kk * SLAB_STRIDE + wl];
      b.y = Cs[(kk + 1) * SLAB_STRIDE + wl];
      const float* wrow = &Wt[(kk >> 1) * 256];
#pragma unroll
      for (int ot = 0; ot < 8; ++ot) {
        v2f a = *(const v2f*)&wrow[(ot * 16 + l) * 2];
        acc[ot] = wmma4(a, b, acc[ot]);
      }
    }
  }
  const int rbase = (lane >> 4) << 3;
#pragma unroll
  for (int ot = 0; ot < 8; ++ot) {
#pragma unroll
    for (int r = 0; r < 8; ++r) {
      int o = ot * 16 + rbase + r;
      size_t idx = (size_t)o * NT_ + j0 + l;
      out[idx] = (acc[ot][r] + gb[o]) * emb[idx] + skip[idx];
    }
  }
}

// ---------------------------------------------------------------------------
extern "C" void kernel_launch(void* const* d_in, const int* in_sizes, int n_in,
                              void* d_out, int out_size, void* d_ws,
                              size_t ws_size, hipStream_t stream) {
  (void)in_sizes; (void)n_in; (void)out_size; (void)ws_size;
  const float* x      = (const float*)d_in[0];
  const float* conv_w = (const float*)d_in[1];
  const float* conv_b = (const float*)d_in[2];
  const float* memory = (const float*)d_in[3];
  const float* fc_w   = (const float*)d_in[4];
  const float* fc_b   = (const float*)d_in[5];
  const float* gcn_w  = (const float*)d_in[6];
  const float* gcn_b  = (const float*)d_in[7];
  const float* emb    = (const float*)d_in[8];
  float* out = (float*)d_out;

  // workspace layout (floats):
  float* ws   = (float*)d_ws;
  float* xc   = ws;                                  // [C, N*T]
  float* cur1 = ws + (size_t)C_ * NT_;               // [C, N*T]
  float* xcs  = ws + 2 * (size_t)C_ * NT_;           // [C, N]
  float* ap1  = xcs + (size_t)C_ * N_;               // [N, N]
  float* ap2  = ap1 + (size_t)N_ * N_;               // [N, N]
  float* adjm = ap2 + (size_t)N_ * N_;               // [N, N]
  float* cur2 = xc;  // xc is dead after k_diffuse #1 -> reuse its buffer

  const size_t lds_big = (128 * 128 + 128 * SLAB_STRIDE) * sizeof(float);

  k_conv   <<<NT_ / 128,     256, lds_big, stream>>>(x, conv_w, conv_b, xc);
  k_sum_t  <<<(C_ * N_) / 8, 256, 0,       stream>>>(xc, xcs);
  k_adj_pre<<<(64 * 64) / 4, 128, 0,       stream>>>(xcs, memory, ap1, ap2);
  k_adj_row<<<N_,             32, 0,       stream>>>(ap1, ap2, fc_w, fc_b, adjm);
  k_diffuse<<<C_ * 16,       256, 0,       stream>>>(xc, adjm, cur1);
  k_diffuse<<<C_ * 16,       256, 0,       stream>>>(cur1, adjm, cur2);
  k_gcn_out<<<NT_ / 128,     256, lds_big, stream>>>(cur1, cur2, gcn_w, gcn_b,
                                                     emb, x, out);
}